// MinimalCortexGPT_16801912062750
// MI455X (gfx1250) — compile-verified
//
#include <hip/hip_runtime.h>
#include <math.h>

// Problem constants (from reference setup_inputs)
#define B_SZ   4096
#define S_SZ   64
#define V_SZ   50257
#define D_SZ   128
#define N_FLAT (B_SZ * D_SZ)        // 524288 query length
#define N_STM  32
#define N_LTM  100
#define N_ROWS (1 + N_STM + N_LTM)  // 133 rows to reduce (row 0 = query itself)
#define NCHUNK 16
#define CHUNK  (N_FLAT / NCHUNK)    // 32768 elems per chunk
#define EPSV   1e-8f

// Workspace layout (float units)
#define WS_X     0
#define WS_PDOT  (WS_X + N_FLAT)
#define WS_PNORM (WS_PDOT + N_ROWS * NCHUNK)
#define WS_CTRL  (WS_PNORM + N_ROWS * NCHUNK)

typedef __attribute__((ext_vector_type(2))) float v2f;
typedef __attribute__((ext_vector_type(8))) float v8f;

// ---------------------------------------------------------------------------
// Kernel 1: x[b][d] = mean_s emb_table[input_ids[b][s]][d]
// 256 threads = 2 rows of B; ids are wave-uniform -> scalar loads.
// ---------------------------------------------------------------------------
__global__ void k_embed_mean(const int* __restrict__ ids,
                             const float* __restrict__ emb,
                             float* __restrict__ x) {
    int tid = threadIdx.x;
    int b = blockIdx.x * 2 + (tid >> 7);
    int d = tid & 127;
    const int* idr = ids + (size_t)b * S_SZ;
    float acc = 0.0f;
#pragma unroll 4
    for (int s = 0; s < S_SZ; ++s) {
        int id = idr[s];
        acc += emb[(size_t)id * D_SZ + d];
    }
    x[(size_t)b * D_SZ + d] = acc * (1.0f / (float)S_SZ);
}

// ---------------------------------------------------------------------------
// Kernel 2: per (row, chunk) partial dot(key,x) and ||key||^2.
// row 0 = x itself; rows 1..32 = stm keys; rows 33..132 = ltm keys.
// Deterministic: each block writes its own partial slot (no atomics).
// ---------------------------------------------------------------------------
__global__ void k_row_partials(const float* __restrict__ x,
                               const float* __restrict__ stm_keys,
                               const float* __restrict__ ltm_keys,
                               float* __restrict__ pdot,
                               float* __restrict__ pnorm) {
    int r = blockIdx.x / NCHUNK;
    int c = blockIdx.x % NCHUNK;
    const float* key;
    if (r == 0)            key = x;
    else if (r <= N_STM)   key = stm_keys + (size_t)(r - 1) * N_FLAT;
    else                   key = ltm_keys + (size_t)(r - 1 - N_STM) * N_FLAT;

    const float4* k4 = (const float4*)key + (size_t)c * (CHUNK / 4);
    const float4* x4 = (const float4*)x   + (size_t)c * (CHUNK / 4);

    float dot = 0.0f, nrm = 0.0f;
    for (int i = threadIdx.x; i < CHUNK / 4; i += 256) {
        float4 kv = k4[i];
        float4 xv = x4[i];
        dot += kv.x * xv.x + kv.y * xv.y + kv.z * xv.z + kv.w * xv.w;
        nrm += kv.x * kv.x + kv.y * kv.y + kv.z * kv.z + kv.w * kv.w;
    }

    __shared__ float sd[256];
    __shared__ float sn[256];
    int tid = threadIdx.x;
    sd[tid] = dot;
    sn[tid] = nrm;
    __syncthreads();
    for (int off = 128; off > 0; off >>= 1) {
        if (tid < off) {
            sd[tid] += sd[tid + off];
            sn[tid] += sn[tid + off];
        }
        __syncthreads();
    }
    if (tid == 0) {
        pdot[blockIdx.x]  = sd[0];
        pnorm[blockIdx.x] = sn[0];
    }
}

// ---------------------------------------------------------------------------
// Kernel 3: finalize — reduce partials, cosine sims, argmax, blend control.
// Single small block; writes stm_conf/ltm_conf to d_out tail.
// ---------------------------------------------------------------------------
__global__ void k_finalize(const float* __restrict__ pdot,
                           const float* __restrict__ pnorm,
                           float* __restrict__ ctrl,
                           float* __restrict__ out_tail) {
    __shared__ float dotv[N_ROWS];
    __shared__ float nrmv[N_ROWS];
    int t = threadIdx.x;
    if (t < N_ROWS) {
        float d = 0.0f, n = 0.0f;
        for (int c = 0; c < NCHUNK; ++c) {
            d += pdot[t * NCHUNK + c];
            n += pnorm[t * NCHUNK + c];
        }
        dotv[t] = d;
        nrmv[t] = n;
    }
    __syncthreads();
    if (t == 0) {
        float qn = fmaxf(sqrtf(nrmv[0]), EPSV);
        float best_stm = -1e30f; int bi_stm = 0;
        for (int m = 0; m < N_STM; ++m) {
            float kn = fmaxf(sqrtf(nrmv[1 + m]), EPSV);
            float sim = dotv[1 + m] / (kn * qn);
            if (sim > best_stm) { best_stm = sim; bi_stm = m; }
        }
        float best_ltm = -1e30f; int bi_ltm = 0;
        for (int m = 0; m < N_LTM; ++m) {
            float kn = fmaxf(sqrtf(nrmv[1 + N_STM + m]), EPSV);
            float sim = dotv[1 + N_STM + m] / (kn * qn);
            if (sim > best_ltm) { best_ltm = sim; bi_ltm = m; }
        }
        out_tail[0] = best_stm;
        out_tail[1] = best_ltm;
        float a = 1.0f, bb = 0.0f; int sel = 0, idx = 0;
        if (best_stm > 0.5f)      { a = 0.5f; bb = 0.5f; sel = 1; idx = bi_stm; }
        else if (best_ltm > 0.5f) { a = 0.7f; bb = 0.3f; sel = 2; idx = bi_ltm; }
        ctrl[0] = a;
        ctrl[1] = bb;
        ctrl[2] = (float)sel;
        ctrl[3] = (float)idx;
    }
}

// ---------------------------------------------------------------------------
// Kernel 4: blend x = a*x + b*val (val selected by ctrl). Elementwise, in place.
// ---------------------------------------------------------------------------
__global__ void k_blend(float* __restrict__ x,
                        const float* __restrict__ stm_vals,
                        const float* __restrict__ ltm_vals,
                        const float* __restrict__ ctrl) {
    float a  = ctrl[0];
    float bb = ctrl[1];
    int  sel = (int)ctrl[2];
    int  idx = (int)ctrl[3];
    const float* val = (sel == 2) ? (ltm_vals + (size_t)idx * N_FLAT)
                                  : (stm_vals + (size_t)idx * N_FLAT);
    size_t i = (size_t)blockIdx.x * 256 + threadIdx.x;   // float4 index
    float4* x4 = (float4*)x;
    const float4* v4 = (const float4*)val;
    float4 xv = x4[i];
    float4 vv = v4[i];
    xv.x = a * xv.x + bb * vv.x;
    xv.y = a * xv.y + bb * vv.y;
    xv.z = a * xv.z + bb * vv.z;
    xv.w = a * xv.w + bb * vv.w;
    x4[i] = xv;
}

// ---------------------------------------------------------------------------
// Kernel 5: logits = x @ out_w^T + out_b using V_WMMA_F32_16X16X4_F32.
// Block = 256 threads (8 wave32s), tile 64(M) x 128(N), K = 128.
// A tile staged in LDS (pitch 132 floats -> conflict-free ds_load_b64).
// Wave w owns 16 N-columns, accumulates 4 stacked 16x16 f32 tiles down M.
//
// V-edge handling: out-of-range lanes clamp their out_w row pointer to a
// valid row and load unconditionally. In 16x16x4 WMMA, column n of D only
// consumes B data from lanes with lane%16 == n, so garbage in tail lanes
// only reaches accumulator elements whose stores are guarded by nvalid.
// This keeps the inner loop free of exec-mask manipulation.
// ---------------------------------------------------------------------------
#define LDS_PITCH 132
__global__ void k_gemm_wmma(const float* __restrict__ x,
                            const float* __restrict__ w,
                            const float* __restrict__ bias,
                            float* __restrict__ out) {
    __shared__ __align__(16) float As[64 * LDS_PITCH];

    int tid = threadIdx.x;
    int m0 = blockIdx.y * 64;
    int n0 = blockIdx.x * 128;

    // Stage A tile: 64 rows x 128 K-cols (2048 float4 loads).
    for (int i = tid; i < 2048; i += 256) {
        int row = i >> 5;       // 32 float4 per row
        int c4  = i & 31;
        float4 v = *(const float4*)(x + (size_t)(m0 + row) * D_SZ + c4 * 4);
        *(float4*)(&As[row * LDS_PITCH + c4 * 4]) = v;
    }
    __syncthreads();

    int lane = tid & 31;
    int wave = tid >> 5;
    int nb = n0 + wave * 16 + (lane & 15);     // this lane's output column
    bool nvalid = nb < V_SZ;
    int nc = nvalid ? nb : (V_SZ - 1);         // clamp: tail lanes read a valid row
    const float* wrow = w + (size_t)nc * D_SZ;
    int khalf = 2 * (lane >> 4);               // lanes 0-15: K pair k0,k0+1; 16-31: k0+2,k0+3

    v8f c0 = {}, c1 = {}, c2 = {}, c3 = {};

#pragma unroll 8
    for (int k0 = 0; k0 < D_SZ; k0 += 4) {
        int kk = k0 + khalf;
        v2f bf = *(const v2f*)(wrow + kk);     // unconditional, 8B aligned
        const float* ab = &As[(lane & 15) * LDS_PITCH + kk];
        v2f a0 = *(const v2f*)(ab);
        v2f a1 = *(const v2f*)(ab + 16 * LDS_PITCH);
        v2f a2 = *(const v2f*)(ab + 32 * LDS_PITCH);
        v2f a3 = *(const v2f*)(ab + 48 * LDS_PITCH);
        c0 = __builtin_amdgcn_wmma_f32_16x16x4_f32(false, a0, false, bf, (short)0, c0, false, false);
        c1 = __builtin_amdgcn_wmma_f32_16x16x4_f32(false, a1, false, bf, (short)0, c1, false, false);
        c2 = __builtin_amdgcn_wmma_f32_16x16x4_f32(false, a2, false, bf, (short)0, c2, false, false);
        c3 = __builtin_amdgcn_wmma_f32_16x16x4_f32(false, a3, false, bf, (short)0, c3, false, false);
    }

    if (nvalid) {
        float bias_n = bias[nb];
        int mloc0 = 8 * (lane >> 4);           // C/D layout: VGPR r holds M = r (+8 for hi lanes)
#pragma unroll
        for (int r = 0; r < 8; ++r) {
            size_t row = (size_t)(m0 + mloc0 + r);
            out[(row +  0) * (size_t)V_SZ + nb] = c0[r] + bias_n;
            out[(row + 16) * (size_t)V_SZ + nb] = c1[r] + bias_n;
            out[(row + 32) * (size_t)V_SZ + nb] = c2[r] + bias_n;
            out[(row + 48) * (size_t)V_SZ + nb] = c3[r] + bias_n;
        }
    }
}

// ---------------------------------------------------------------------------
extern "C" void kernel_launch(void* const* d_in, const int* in_sizes, int n_in,
                              void* d_out, int out_size, void* d_ws, size_t ws_size,
                              hipStream_t stream) {
    const int*   ids   = (const int*)d_in[0];
    const float* emb   = (const float*)d_in[1];
    const float* out_w = (const float*)d_in[2];
    const float* out_b = (const float*)d_in[3];
    const float* stm_k = (const float*)d_in[4];
    const float* stm_v = (const float*)d_in[5];
    const float* ltm_k = (const float*)d_in[6];
    const float* ltm_v = (const float*)d_in[7];
    float* out = (float*)d_out;
    float* ws  = (float*)d_ws;

    float* x     = ws + WS_X;
    float* pdot  = ws + WS_PDOT;
    float* pnorm = ws + WS_PNORM;
    float* ctrl  = ws + WS_CTRL;

    // 1) embed + mean pool
    k_embed_mean<<<B_SZ / 2, 256, 0, stream>>>(ids, emb, x);
    // 2) dot / norm partials for all 133 rows
    k_row_partials<<<N_ROWS * NCHUNK, 256, 0, stream>>>(x, stm_k, ltm_k, pdot, pnorm);
    // 3) cosine sims, argmax, confs, blend control
    k_finalize<<<1, 160, 0, stream>>>(pdot, pnorm, ctrl, out + (size_t)B_SZ * V_SZ);
    // 4) blend x in place
    k_blend<<<N_FLAT / 4 / 256, 256, 0, stream>>>(x, stm_v, ltm_v, ctrl);
    // 5) WMMA GEMM: logits = x @ out_w^T + out_b
    dim3 grid((V_SZ + 127) / 128, B_SZ / 64);
    k_gemm_wmma<<<grid, 256, 0, stream>>>(x, out_w, out_b, out);
}